// Dgcn_55825984914101
// MI455X (gfx1250) — compile-verified
//
#include <hip/hip_runtime.h>
#include <hip/hip_bf16.h>
#include <math.h>

// ---- problem constants -----------------------------------------------------
#define NB   64            // batch
#define NN   62            // nodes
#define NCH  4             // C
#define NT   2048          // T
#define NK   3             // K (cheb orders)
#define ND   10            // D
#define RC   248           // N*C  (GEMM M and K dims)
#define RCP  256           // padded row count of W storage
#define EPS  1e-5f

typedef __attribute__((ext_vector_type(2))) float v2f;
typedef __attribute__((ext_vector_type(8))) float v8f;

// single-instruction relu: v_med3_f32(x, 0, +inf), no canonicalize needed
__device__ __forceinline__ float relu1(float x) {
    return __builtin_amdgcn_fmed3f(x, 0.f, __builtin_inff());
}

// ---------------------------------------------------------------------------
// Kernel 0: fold cheb / adaptive adjacency / alpha / Theta into W[256][248]
// and zero the BN stat accumulators (512 floats).
// ---------------------------------------------------------------------------
__global__ void prep_kernel(const float* __restrict__ cheb,   // [K][N][N]
                            const float* __restrict__ nv1,    // [N][D]
                            const float* __restrict__ nv2,    // [D][N]
                            const float* __restrict__ alpha,  // [K][2]
                            const float* __restrict__ theta,  // [K][C][4]
                            float* __restrict__ W,            // [256][248]
                            float* __restrict__ stats)        // [512]
{
    __shared__ float sAdp[NN][NN];   // adp[n][m], softmax over m
    __shared__ float sAb[NK][2];

    const int tid = threadIdx.x;
    stats[tid]       = 0.f;          // blockDim.x == 256
    stats[tid + 256] = 0.f;

    if (tid < NN) {                  // thread tid owns adjacency row n = tid
        float mx = 0.f;
        for (int m = 0; m < NN; ++m) {
            float d = 0.f;
            for (int e = 0; e < ND; ++e) d += nv1[tid * ND + e] * nv2[e * NN + m];
            d = fmaxf(d, 0.f);
            sAdp[tid][m] = d;
            mx = fmaxf(mx, d);
        }
        float s = 0.f;
        if (mx > 0.f) {
            for (int m = 0; m < NN; ++m) {
                float d = sAdp[tid][m];
                float p = (d > 0.f) ? __expf(d - mx) : 0.f;  // masked == -1e10 -> 0
                sAdp[tid][m] = p;
                s += p;
            }
        } else {                     // entire row masked -> uniform softmax
            for (int m = 0; m < NN; ++m) sAdp[tid][m] = 1.f;
            s = (float)NN;
        }
        const float inv = 1.f / s;
        for (int m = 0; m < NN; ++m) sAdp[tid][m] *= inv;
    }
    if (tid < NK) {
        float e0 = __expf(alpha[tid * 2 + 0]);
        float e1 = __expf(alpha[tid * 2 + 1]);
        float inv = 1.f / (e0 + e1);
        sAb[tid][0] = e0 * inv;
        sAb[tid][1] = e1 * inv;
    }
    __syncthreads();

    for (int e = tid; e < RCP * RC; e += blockDim.x) {
        const int row = e / RC;      // r = m*4 + j
        const int col = e % RC;      // s = n*4 + c
        float w = 0.f;
        if (row < RC) {
            const int m = row >> 2, j = row & 3;
            const int n = col >> 2, c = col & 3;
            #pragma unroll
            for (int k = 0; k < NK; ++k) {
                float tk = sAb[k][0] * cheb[(k * NN + n) * NN + m]
                         + sAb[k][1] * sAdp[n][m];
                w += tk * theta[(k * NCH + c) * NCH + j];
            }
        }
        W[e] = w;
    }
}

// ---------------------------------------------------------------------------
// Kernel 1: per-batch GEMM  out[b] = relu( W[248x248] @ relu(x[b]) ) via
// v_wmma_f32_16x16x4_f32, plus per-row sum / sum-of-squares for BatchNorm.
// Block = 256 threads (8 waves), covering (batch b) x (256-col strip).
// Wave w owns row tiles at [32w, 32w+16); wave 7's second tile is re-based
// to row 232 (overlapping rows 232..239) so every store is unconditional.
// x loads are non-temporal (streamed once) so the pre-BN output stays
// resident in the 192MB L2 for the BN pass.
// ---------------------------------------------------------------------------
__global__ __launch_bounds__(256) void gemm_kernel(
    const float* __restrict__ x,     // [B][248][2048]
    const float* __restrict__ W,     // [256][248]
    float* __restrict__ out,         // [B][248][2048] (pre-BN, relu'd)
    float* __restrict__ stats)       // [512]: rowSum[256], rowSumSq[256]
{
    const int b    = blockIdx.y;
    const int ts   = blockIdx.x * 256;          // column-strip base in T
    const int wave = threadIdx.x >> 5;
    const int lane = threadIdx.x & 31;
    const int half = lane >> 4;                 // 0: lanes 0-15, 1: lanes 16-31
    const int tl   = lane & 15;

    const int rb0 = wave * 32;                  // first row tile
    const int rb1 = (wave == 7) ? 232 : (rb0 + 16);   // second (overlap trick)

    const float* __restrict__ xb = x   + (size_t)b * RC * NT;
    float*       __restrict__ ob = out + (size_t)b * RC * NT;

    // per-(rowtile,i) running stats; the output row for a given (rt,i,half)
    // is fixed across all column tiles, so accumulate in registers.
    float sumv[16], sumq[16];
    #pragma unroll
    for (int i = 0; i < 16; ++i) { sumv[i] = 0.f; sumq[i] = 0.f; }

    for (int cb = 0; cb < 4; ++cb) {            // 4 x (4 col tiles) = 256 cols
        const int t0 = ts + cb * 64 + tl;       // this lane's column (+cc*16)

        v8f acc[2][4];
        #pragma unroll
        for (int rt = 0; rt < 2; ++rt)
            #pragma unroll
            for (int cc = 0; cc < 4; ++cc)
                acc[rt][cc] = (v8f){0.f,0.f,0.f,0.f,0.f,0.f,0.f,0.f};

        for (int k0 = 0; k0 < RC; k0 += 4) {
            // A fragments (16x4 f32): lanes 0-15 -> K = k0,k0+1 ; 16-31 -> k0+2,k0+3
            const v2f a0 = *(const v2f*)(W + (size_t)(rb0 + tl) * RC + k0 + 2 * half);
            const v2f a1 = *(const v2f*)(W + (size_t)(rb1 + tl) * RC + k0 + 2 * half);

            const float* xp = xb + (size_t)(k0 + 2 * half) * NT + t0;
            #pragma unroll
            for (int cc = 0; cc < 4; ++cc) {
                // B fragment (4x16 f32): v0 = row k0(+2h), v1 = row k0+1(+2h)
                v2f bf;
                bf.x = relu1(__builtin_nontemporal_load(xp + cc * 16));
                bf.y = relu1(__builtin_nontemporal_load(xp + NT + cc * 16));
                acc[0][cc] = __builtin_amdgcn_wmma_f32_16x16x4_f32(
                    false, a0, false, bf, (short)0, acc[0][cc], false, false);
                acc[1][cc] = __builtin_amdgcn_wmma_f32_16x16x4_f32(
                    false, a1, false, bf, (short)0, acc[1][cc], false, false);
            }
        }

        // epilogue: relu, store (unconditional, immediate offsets), stats.
        // D layout: VGPR i -> row (i + 8*half), col = tl within the tile.
        #pragma unroll
        for (int rt = 0; rt < 2; ++rt) {
            const int rb = (rt == 0) ? rb0 : rb1;
            float* __restrict__ p = ob + (size_t)(rb + 8 * half) * NT + t0;
            #pragma unroll
            for (int cc = 0; cc < 4; ++cc) {
                #pragma unroll
                for (int i = 0; i < 8; ++i) {
                    float v = relu1(acc[rt][cc][i]);
                    p[(size_t)i * NT + cc * 16] = v;   // const byte offsets
                    sumv[rt * 8 + i] += v;
                    sumq[rt * 8 + i] += v * v;
                }
            }
        }
    }

    // reduce each running stat across the 16 lanes of its half (cols), then
    // one global atomic per row per block.  Rows 232..239 were computed twice
    // for wave 7 (rt1/half0 duplicates rt0/half1) -> skip the duplicate.
    #pragma unroll
    for (int i = 0; i < 16; ++i) {
        float s = sumv[i], q = sumq[i];
        #pragma unroll
        for (int m = 1; m < 16; m <<= 1) {
            s += __shfl_xor(s, m, 32);
            q += __shfl_xor(q, m, 32);
        }
        const bool isRt1 = (i >= 8);
        const bool dup   = (wave == 7) && isRt1 && (half == 0);
        if (tl == 0 && !dup) {
            const int row = (isRt1 ? rb1 : rb0) + (i & 7) + 8 * half;
            atomicAdd(&stats[row],       s);
            atomicAdd(&stats[256 + row], q);
        }
    }
}

// ---------------------------------------------------------------------------
// Kernel 2: collapse row sums into per-channel mean / rstd (62 channels).
// ---------------------------------------------------------------------------
__global__ void finalize_stats(const float* __restrict__ stats,  // [512]
                               float* __restrict__ ch)           // [128]: mean,rstd
{
    const int m = threadIdx.x;
    if (m < NN) {
        const float inv_cnt = 1.f / (float)(NB * NCH * NT);   // 1/524288
        float s = stats[4 * m] + stats[4 * m + 1] + stats[4 * m + 2] + stats[4 * m + 3];
        float q = stats[256 + 4 * m] + stats[256 + 4 * m + 1]
                + stats[256 + 4 * m + 2] + stats[256 + 4 * m + 3];
        const float mean = s * inv_cnt;
        const float var  = q * inv_cnt - mean * mean;          // biased
        ch[m]      = mean;
        ch[64 + m] = rsqrtf(var + EPS);
    }
}

// ---------------------------------------------------------------------------
// Kernel 3: BatchNorm2d (affine=False, training stats) in place over d_out.
// ---------------------------------------------------------------------------
__global__ void bn_kernel(float* __restrict__ out, const float* __restrict__ ch)
{
    const size_t total4 = (size_t)NB * RC * NT / 4;
    size_t idx = (size_t)blockIdx.x * blockDim.x + threadIdx.x;
    if (idx >= total4) return;

    const size_t flat = idx * 4;
    const int row = (int)((flat >> 11) % RC);             // NT == 2048 == 1<<11
    const int m   = row >> 2;

    const float mean = ch[m];
    const float rstd = ch[64 + m];

    float4 v = *(float4*)(out + flat);
    v.x = (v.x - mean) * rstd;
    v.y = (v.y - mean) * rstd;
    v.z = (v.z - mean) * rstd;
    v.w = (v.w - mean) * rstd;
    *(float4*)(out + flat) = v;
}

// ---------------------------------------------------------------------------
extern "C" void kernel_launch(void* const* d_in, const int* in_sizes, int n_in,
                              void* d_out, int out_size, void* d_ws, size_t ws_size,
                              hipStream_t stream) {
    (void)in_sizes; (void)n_in; (void)out_size; (void)ws_size;
    const float* x     = (const float*)d_in[0];
    const float* cheb  = (const float*)d_in[1];
    const float* nv1   = (const float*)d_in[2];
    const float* nv2   = (const float*)d_in[3];
    const float* alpha = (const float*)d_in[4];
    const float* theta = (const float*)d_in[5];
    float* out = (float*)d_out;

    float* W     = (float*)d_ws;         // 256*248 floats = 253952 B
    float* stats = W + RCP * RC;         // 512 floats
    float* ch    = stats + 512;          // 128 floats: mean[64], rstd[64]

    prep_kernel<<<1, 256, 0, stream>>>(cheb, nv1, nv2, alpha, theta, W, stats);

    dim3 grid(NT / 256, NB);             // (8, 64)
    gemm_kernel<<<grid, 256, 0, stream>>>(x, W, out, stats);

    finalize_stats<<<1, 64, 0, stream>>>(stats, ch);

    const size_t total4 = (size_t)NB * RC * NT / 4;
    bn_kernel<<<(unsigned)((total4 + 255) / 256), 256, 0, stream>>>(out, ch);
}